// HarmonicSynthesizer_77206332113005
// MI455X (gfx1250) — compile-verified
//
#include <hip/hip_runtime.h>
#include <math.h>

// ---------------- problem constants ----------------
#define SR_F     24000.0f
#define T_FR     400            // frames
#define NB       4              // batch
#define NH       100            // harmonics
#define L_AUD    96000          // T_FR * 240
#define NCHUNK   375            // L_AUD / 256
#define CHUNK    256
#define TWO_PI_F 6.28318530717958647692f

typedef __attribute__((ext_vector_type(2))) float v2f;
typedef __attribute__((ext_vector_type(8))) float v8f;

// ---------------- workspace layout (in floats) ----------------
#define WS_H    0          // h buffer: 4*32*400 = 51200
#define WS_MIX  51200      // mix: 4*400 = 1600
#define WS_CS   52800      // chunk sums: 4*375 = 1500
#define WS_CB   54300      // chunk bases: 4*375 = 1500

// ---------------- helpers ----------------
__device__ __forceinline__ void interp_coef(int l, int& lo, int& hi, float& w) {
    // F.interpolate(linear, align_corners=False): pos = (l+0.5)*T/L - 0.5
    float pos = ((float)l + 0.5f) * ((float)T_FR / (float)L_AUD) - 0.5f;
    pos = fminf(fmaxf(pos, 0.0f), (float)(T_FR - 1));
    lo = (int)floorf(pos);
    hi = lo + 1; if (hi > T_FR - 1) hi = T_FR - 1;
    w = pos - (float)lo;
}

__device__ __forceinline__ float phase_step(const float* __restrict__ f0b, int l) {
    int lo, hi; float w;
    interp_coef(l, lo, hi, w);
    float f = f0b[lo] * (1.0f - w) + f0b[hi] * w;
    return TWO_PI_F * f / SR_F;
}

// ---------------- conv1 as fp32 WMMA GEMM ----------------
// Per batch: D[32 x 400] = W1[32 x 384] * X[384 x 400] (+bias, leaky_relu)
// Tap-major K ordering: k = tap*128 + c  =>
//   A[m][k] = w1[m*384 + c*3 + tap]
//   B[k][n] = condition[b, c, tcol + tap - 1] (zero pad, branchless)
// One wave per 16x16 output tile; 3 x 32 = 96 V_WMMA_F32_16X16X4_F32 steps.
__global__ void conv1_wmma_kernel(const float* __restrict__ cond,
                                  const float* __restrict__ w1,
                                  const float* __restrict__ b1,
                                  float* __restrict__ hout) {
    const int lane   = threadIdx.x;      // 0..31, full wave32 (EXEC all ones)
    const int nT     = blockIdx.x;       // 0..24 time tile
    const int mT     = blockIdx.y;       // 0..1  out-channel tile
    const int bb     = blockIdx.z;       // 0..3  batch
    const int half   = lane >> 4;        // 0: K-pair {0,1}  1: K-pair {2,3}
    const int lane16 = lane & 15;
    const int tcol   = nT * 16 + lane16; // time column for B/D

    const float* wrow  = w1 + (size_t)(mT * 16 + lane16) * 384;  // A row
    const float* cbase = cond + (size_t)bb * 128 * T_FR;

    v8f acc = {0.f, 0.f, 0.f, 0.f, 0.f, 0.f, 0.f, 0.f};

#pragma unroll
    for (int tap = 0; tap < 3; ++tap) {
        // pad handling hoisted out of the K loop: clamp address, scale by 0/1
        const int   tt    = tcol + tap - 1;
        const float maskf = ((unsigned)tt < (unsigned)T_FR) ? 1.0f : 0.0f;
        int ttc = tt; if (ttc < 0) ttc = 0; if (ttc > T_FR - 1) ttc = T_FR - 1;
        const float* cb2 = cbase + ttc;
        const float* wt  = wrow + tap;

        for (int c0 = 0; c0 < 128; c0 += 4) {
            const int ca = c0 + half * 2;     // this lane's channel pair
            v2f a, bv;
            a.x  = wt[ca * 3];
            a.y  = wt[ca * 3 + 3];
            bv.x = cb2[ca * T_FR] * maskf;
            bv.y = cb2[(ca + 1) * T_FR] * maskf;
            acc = __builtin_amdgcn_wmma_f32_16x16x4_f32(
                      /*neg_a=*/false, a, /*neg_b=*/false, bv,
                      /*c_mod=*/(short)0, acc, /*reuse_a=*/false, /*reuse_b=*/false);
        }
    }

    // D layout: VGPR v, lanes 0-15 -> M=v, lanes 16-31 -> M=v+8 ; N = lane16
    float* hb = hout + (size_t)bb * 32 * T_FR;
#pragma unroll
    for (int v = 0; v < 8; ++v) {
        int m = mT * 16 + v + half * 8;
        float val = acc[v] + b1[m];
        val = (val > 0.0f) ? val : 0.1f * val;   // leaky_relu(0.1)
        hb[m * T_FR + tcol] = val;
    }
}

// ---------------- conv2 (32 -> 1) + sigmoid ----------------
__global__ void conv2_mix_kernel(const float* __restrict__ hin,
                                 const float* __restrict__ w2,
                                 const float* __restrict__ b2,
                                 float* __restrict__ mix) {
    int idx = blockIdx.x * blockDim.x + threadIdx.x;
    if (idx >= NB * T_FR) return;
    int bb = idx / T_FR, t = idx % T_FR;
    const float* hb = hin + (size_t)bb * 32 * T_FR;
    float acc = b2[0];
#pragma unroll
    for (int tap = 0; tap < 3; ++tap) {
        int tt = t + tap - 1;
        float maskf = ((unsigned)tt < (unsigned)T_FR) ? 1.0f : 0.0f;
        int ttc = tt; if (ttc < 0) ttc = 0; if (ttc > T_FR - 1) ttc = T_FR - 1;
        for (int c = 0; c < 32; ++c)
            acc += hb[c * T_FR + ttc] * maskf * w2[c * 3 + tap];
    }
    mix[idx] = 1.0f / (1.0f + __expf(-acc));
}

// ---------------- phase scan, level 1: per-chunk sums ----------------
__global__ void chunk_sum_kernel(const float* __restrict__ f0,
                                 float* __restrict__ csum) {
    __shared__ float red[CHUNK];
    const int chunk = blockIdx.x, bb = blockIdx.y, tid = threadIdx.x;
    const int l = chunk * CHUNK + tid;
    const float* f0b = f0 + bb * T_FR;
    red[tid] = (l < L_AUD) ? phase_step(f0b, l) : 0.0f;
    __syncthreads();
    for (int o = CHUNK / 2; o > 0; o >>= 1) {
        if (tid < o) red[tid] += red[tid + o];
        __syncthreads();
    }
    if (tid == 0) csum[bb * NCHUNK + chunk] = red[0];
}

// ---------------- phase scan, level 2: per-batch exclusive scan ----------------
__global__ void chunk_scan_kernel(const float* __restrict__ csum,
                                  float* __restrict__ cbase) {
    const int bb = threadIdx.x;
    if (bb < NB) {
        float run = 0.0f;
        for (int i = 0; i < NCHUNK; ++i) {
            cbase[bb * NCHUNK + i] = run;
            run += csum[bb * NCHUNK + i];
        }
    }
}

// ---------------- fused: block scan + harmonic synthesis + mix ----------------
__global__ void synth_kernel(const float* __restrict__ f0,
                             const float* __restrict__ amp,
                             const float* __restrict__ offs,
                             const float* __restrict__ rst,
                             const float* __restrict__ pat,
                             const float* __restrict__ noise,
                             const float* __restrict__ mix,
                             const float* __restrict__ cbase,
                             float* __restrict__ out) {
    __shared__ float sc[CHUNK];
    __shared__ float ampS[NH * 3];
    __shared__ float offS[NH * 3];
    __shared__ float patS[NH * 3];

    const int chunk = blockIdx.x, bb = blockIdx.y, tid = threadIdx.x;
    const int l = chunk * CHUNK + tid;
    const float* f0b = f0 + bb * T_FR;

    // inclusive scan of phase step within chunk (Hillis-Steele)
    sc[tid] = phase_step(f0b, l);
    __syncthreads();
    for (int o = 1; o < CHUNK; o <<= 1) {
        float v = (tid >= o) ? sc[tid - o] : 0.0f;
        __syncthreads();
        sc[tid] += v;
        __syncthreads();
    }
    const float phase = cbase[bb * NCHUNK + chunk] + sc[tid];

    // a 256-sample chunk touches at most 3 consecutive frames -> stage in LDS
    float pos0 = ((float)(chunk * CHUNK) + 0.5f) * ((float)T_FR / (float)L_AUD) - 0.5f;
    int fbase = (int)floorf(fmaxf(pos0, 0.0f));
    if (fbase > T_FR - 1) fbase = T_FR - 1;

    const float* ampB = amp  + (size_t)bb * NH * T_FR;
    const float* offB = offs + (size_t)bb * NH * T_FR;
    const float* patB = pat  + (size_t)bb * NH * T_FR;
    for (int i = tid; i < NH * 3; i += CHUNK) {
        int hh = i / 3, j = i % 3;
        int fr = fbase + j; if (fr > T_FR - 1) fr = T_FR - 1;
        ampS[i] = ampB[hh * T_FR + fr];
        offS[i] = offB[hh * T_FR + fr];
        patS[i] = patB[hh * T_FR + fr];
    }
    __syncthreads();

    int lo, hi; float w;
    interp_coef(l, lo, hi, w);
    const float wm = 1.0f - w;
    int jlo = lo - fbase; jlo = (jlo < 0) ? 0 : ((jlo > 2) ? 2 : jlo);
    int jhi = hi - fbase; jhi = (jhi < 0) ? 0 : ((jhi > 2) ? 2 : jhi);

    const float* rb = rst + bb * T_FR;
    const float rv  = rb[lo] * wm + rb[hi] * w;
    const float omr = 1.0f - rv;

    // hoist LDS base pointers: per-harmonic addressing is a +3 increment
    const float* aLo = ampS + jlo; const float* aHi = ampS + jhi;
    const float* oLo = offS + jlo; const float* oHi = offS + jhi;
    const float* pLo = patS + jlo; const float* pHi = patS + jhi;

    float acc = 0.0f;
    for (int hh = 0; hh < NH; ++hh) {
        const int i3 = hh * 3;
        float a  = aLo[i3] * wm + aHi[i3] * w;
        float o2 = oLo[i3] * wm + oHi[i3] * w;
        float p2 = pLo[i3] * wm + pHi[i3] * w;
        float fp = (phase * (float)(hh + 1) + o2) * omr + p2 * rv;
        acc += __sinf(fp) * a;      // v_sin_f32 (TRANS, co-executes with VALU)
    }

    const float* mb = mix + bb * T_FR;
    float mv = mb[lo] * wm + mb[hi] * w;
    float fm = 0.2f * mv;           // NOISE_RATIO * mix_up
    out[(size_t)bb * L_AUD + l] =
        (1.0f - fm) * acc + fm * noise[(size_t)bb * L_AUD + l];
}

// ---------------- launch ----------------
extern "C" void kernel_launch(void* const* d_in, const int* in_sizes, int n_in,
                              void* d_out, int out_size, void* d_ws, size_t ws_size,
                              hipStream_t stream) {
    const float* f0    = (const float*)d_in[0];
    const float* cond  = (const float*)d_in[1];
    const float* amp   = (const float*)d_in[2];
    const float* offs  = (const float*)d_in[3];
    const float* rst   = (const float*)d_in[4];
    const float* pat   = (const float*)d_in[5];
    const float* noise = (const float*)d_in[6];
    const float* w1    = (const float*)d_in[7];
    const float* b1    = (const float*)d_in[8];
    const float* w2    = (const float*)d_in[9];
    const float* b2    = (const float*)d_in[10];
    float* out = (float*)d_out;

    float* W      = (float*)d_ws;
    float* hbuf   = W + WS_H;
    float* mixbuf = W + WS_MIX;
    float* csum   = W + WS_CS;
    float* cb     = W + WS_CB;

    // conv1: fp32 WMMA GEMM, one wave per 16x16 tile
    conv1_wmma_kernel<<<dim3(25, 2, NB), 32, 0, stream>>>(cond, w1, b1, hbuf);
    // conv2 + sigmoid
    conv2_mix_kernel<<<dim3((NB * T_FR + 255) / 256), 256, 0, stream>>>(hbuf, w2, b2, mixbuf);
    // hierarchical phase cumsum
    chunk_sum_kernel<<<dim3(NCHUNK, NB), CHUNK, 0, stream>>>(f0, csum);
    chunk_scan_kernel<<<1, 32, 0, stream>>>(csum, cb);
    // fused scan + harmonic synthesis + noise mix
    synth_kernel<<<dim3(NCHUNK, NB), CHUNK, 0, stream>>>(f0, amp, offs, rst, pat,
                                                         noise, mixbuf, cb, out);
}